// SelfAttn_12197707120922
// MI455X (gfx1250) — compile-verified
//
#include <hip/hip_runtime.h>
#include <math.h>

typedef __attribute__((ext_vector_type(2))) float v2f;
typedef __attribute__((ext_vector_type(8))) float v8f;

#define DIM   512
#define SEQL  1024
#define NSEQ  256

#define FLASH_THREADS 512
#define FLASH_WAVES   (FLASH_THREADS / 32)

// ---------------------------------------------------------------------------
// Kernel 1: QW[n,k] = sum_d Q[n,d] * W[d,k], Q[n,:] = input_seq[n, L-1, :]
// One wave (32 threads) per 16x16 output tile, fp32 WMMA, K stepped by 4.
// A layout (16x4 f32): lane half selects K pair {0,1}/{2,3}; lane&15 = row M.
// B layout (4x16 f32): lane half selects K pair; lane&15 = col N.
// C/D layout: VGPR r holds row r (+8 for upper lane half), col = lane&15.
// ---------------------------------------------------------------------------
__global__ __launch_bounds__(32) void qw_gemm_wmma(
    const float* __restrict__ X,   // input_seq [N, L, D]
    const float* __restrict__ W,   // [D, D]
    float* __restrict__ QW)        // [N, D]
{
    const int ntile = blockIdx.x;        // 0..31  (columns of QW)
    const int mtile = blockIdx.y;        // 0..15  (rows of QW = n)
    const int lane  = threadIdx.x;
    const int half  = lane >> 4;
    const int l15   = lane & 15;
    const int kb    = half * 2;

    const float* qrow = X + ((size_t)(mtile * 16 + l15) * SEQL + (SEQL - 1)) * DIM;
    const int    ncol = ntile * 16 + l15;

    v8f c = {};
    for (int k = 0; k < DIM; k += 4) {
        v2f a = *(const v2f*)(qrow + k + kb);            // A[m][k+kb], A[m][k+kb+1]
        v2f b;
        b.x = W[(size_t)(k + kb)     * DIM + ncol];      // B[k+kb  ][n]
        b.y = W[(size_t)(k + kb + 1) * DIM + ncol];      // B[k+kb+1][n]
        c = __builtin_amdgcn_wmma_f32_16x16x4_f32(false, a, false, b,
                                                  (short)0, c, false, false);
    }
#pragma unroll
    for (int r = 0; r < 8; ++r) {
        const int mrow = mtile * 16 + half * 8 + r;
        QW[(size_t)mrow * DIM + ncol] = c[r];
    }
}

// ---------------------------------------------------------------------------
// Kernel 2: single-pass online-softmax pooling (the 512 MB streaming pass).
// One block (16 waves) per sequence n. Each wave owns rows l = wave + 16*i,
// keeps private (m, Z, acc[16 floats/lane]); waves combined via LDS at end.
// XBAR[n,k] = (sum_l softmax_l * x[n,l,k])   (already normalized by Z)
// ---------------------------------------------------------------------------
__global__ __launch_bounds__(FLASH_THREADS) void flash_pool(
    const float* __restrict__ X,    // input_seq [N, L, D]
    const float* __restrict__ QW,   // [N, D]
    float* __restrict__ XBAR)       // [N, D]
{
    __shared__ float s_acc[FLASH_WAVES][DIM];
    __shared__ float s_m[FLASH_WAVES];
    __shared__ float s_z[FLASH_WAVES];

    const int n    = blockIdx.x;
    const int tid  = threadIdx.x;
    const int wave = tid >> 5;
    const int lane = tid & 31;

    const float* xbase = X + (size_t)n * SEQL * DIM;
    const float* qw    = QW + (size_t)n * DIM;

    float4 q4[4], acc[4];
#pragma unroll
    for (int j = 0; j < 4; ++j) {
        q4[j]  = *(const float4*)(qw + lane * 4 + 128 * j);
        acc[j] = make_float4(0.f, 0.f, 0.f, 0.f);
    }
    float m = -INFINITY, Z = 0.f;

    for (int l = wave; l < SEQL; l += FLASH_WAVES) {
        const float* xrow = xbase + (size_t)l * DIM;
        float4 x4[4];
        float  s = 0.f;
#pragma unroll
        for (int j = 0; j < 4; ++j) {
            x4[j] = *(const float4*)(xrow + lane * 4 + 128 * j);
            s += q4[j].x * x4[j].x + q4[j].y * x4[j].y +
                 q4[j].z * x4[j].z + q4[j].w * x4[j].w;
        }
        // wave32 reduction -> every lane holds the full dot product
#pragma unroll
        for (int off = 16; off >= 1; off >>= 1)
            s += __shfl_xor(s, off, 32);

        // branchless online softmax update
        const float m_new = fmaxf(m, s);
        const float scale = __expf(m - m_new);   // 1 if m unchanged, 0 on first row
        const float w     = __expf(s - m_new);
        Z = Z * scale + w;
#pragma unroll
        for (int j = 0; j < 4; ++j) {
            acc[j].x = acc[j].x * scale + w * x4[j].x;
            acc[j].y = acc[j].y * scale + w * x4[j].y;
            acc[j].z = acc[j].z * scale + w * x4[j].z;
            acc[j].w = acc[j].w * scale + w * x4[j].w;
        }
        m = m_new;
    }

#pragma unroll
    for (int j = 0; j < 4; ++j)
        *(float4*)(&s_acc[wave][lane * 4 + 128 * j]) = acc[j];
    if (lane == 0) { s_m[wave] = m; s_z[wave] = Z; }
    __syncthreads();

    // cross-wave combine: 512 threads, one column each
    const int c = tid;
    float mg = -INFINITY;
#pragma unroll
    for (int i = 0; i < FLASH_WAVES; ++i) mg = fmaxf(mg, s_m[i]);
    float zg = 0.f, num = 0.f;
#pragma unroll
    for (int i = 0; i < FLASH_WAVES; ++i) {
        const float e = __expf(s_m[i] - mg);
        zg  += s_z[i] * e;
        num += s_acc[i][c] * e;
    }
    XBAR[(size_t)n * DIM + c] = num / zg;
}

// ---------------------------------------------------------------------------
// Kernel 3: P = XBAR @ V^T  (P[n,d] = sum_k V[d,k] * XBAR[n,k]), fp32 WMMA.
// B operand element (k, ncol) = V[ncol][k] -> contiguous v2f load from V row.
// ---------------------------------------------------------------------------
__global__ __launch_bounds__(32) void pooled_gemm_wmma(
    const float* __restrict__ XBAR,  // [N, D]
    const float* __restrict__ V,     // [D, D]
    float* __restrict__ P)           // [N, D]
{
    const int ntile = blockIdx.x;
    const int mtile = blockIdx.y;
    const int lane  = threadIdx.x;
    const int half  = lane >> 4;
    const int l15   = lane & 15;
    const int kb    = half * 2;

    const float* arow = XBAR + (size_t)(mtile * 16 + l15) * DIM;
    const float* vrow = V + (size_t)(ntile * 16 + l15) * DIM;  // row ncol of V

    v8f c = {};
    for (int k = 0; k < DIM; k += 4) {
        v2f a = *(const v2f*)(arow + k + kb);
        v2f b = *(const v2f*)(vrow + k + kb);   // V[ncol][k+kb], V[ncol][k+kb+1]
        c = __builtin_amdgcn_wmma_f32_16x16x4_f32(false, a, false, b,
                                                  (short)0, c, false, false);
    }
#pragma unroll
    for (int r = 0; r < 8; ++r) {
        const int mrow = mtile * 16 + half * 8 + r;
        P[(size_t)mrow * DIM + ntile * 16 + l15] = c[r];
    }
}

// ---------------------------------------------------------------------------
// Kernel 4: out[n] = sum_d relu(P[n,d]) * v[d]
// ---------------------------------------------------------------------------
__global__ __launch_bounds__(128) void out_reduce(
    const float* __restrict__ P,
    const float* __restrict__ v,
    float* __restrict__ out)
{
    __shared__ float red[4];
    const int n   = blockIdx.x;
    const int tid = threadIdx.x;
    const float* p = P + (size_t)n * DIM;

    float acc = 0.f;
#pragma unroll
    for (int j = 0; j < 4; ++j) {
        const int c = tid + 128 * j;
        acc += fmaxf(p[c], 0.f) * v[c];
    }
#pragma unroll
    for (int off = 16; off >= 1; off >>= 1)
        acc += __shfl_xor(acc, off, 32);
    if ((tid & 31) == 0) red[tid >> 5] = acc;
    __syncthreads();
    if (tid == 0) out[n] = red[0] + red[1] + red[2] + red[3];
}

// ---------------------------------------------------------------------------
extern "C" void kernel_launch(void* const* d_in, const int* in_sizes, int n_in,
                              void* d_out, int out_size, void* d_ws, size_t ws_size,
                              hipStream_t stream) {
    (void)in_sizes; (void)n_in; (void)out_size; (void)ws_size;

    const float* X  = (const float*)d_in[0];   // input_seq [256,1024,512]
    const float* W  = (const float*)d_in[1];   // [512,512]
    const float* Vm = (const float*)d_in[2];   // [512,512]
    const float* vv = (const float*)d_in[3];   // [512]
    float*       out = (float*)d_out;          // [256]

    float* QW   = (float*)d_ws;                // 256*512 f32
    float* XBAR = QW   + (size_t)NSEQ * DIM;   // 256*512 f32
    float* P    = XBAR + (size_t)NSEQ * DIM;   // 256*512 f32

    qw_gemm_wmma<<<dim3(DIM / 16, NSEQ / 16), 32, 0, stream>>>(X, W, QW);
    flash_pool<<<NSEQ, FLASH_THREADS, 0, stream>>>(X, QW, XBAR);
    pooled_gemm_wmma<<<dim3(DIM / 16, NSEQ / 16), 32, 0, stream>>>(XBAR, Vm, P);
    out_reduce<<<NSEQ, 128, 0, stream>>>(P, vv, out);
}